// ContrastiveLoss_31945966748219
// MI455X (gfx1250) — compile-verified
//
#include <hip/hip_runtime.h>
#include <math.h>

#define NUM_CLASSES 10
#define BATCH 2048
#define DIM 128
#define TEMP 0.07f
#define BASE_TEMP 0.07f

#define MTILE 32
#define NTILE 64
#define NSTEPS (BATCH / NTILE)       // 32
#define LDS_STRIDE 132               // 128 + 4 dword pad -> conflict-free strided K reads
#define NBLOCKS (NUM_CLASSES * (BATCH / MTILE))  // 640

typedef __attribute__((ext_vector_type(2))) float v2f;
typedef __attribute__((ext_vector_type(8))) float v8f;

// ---------------- Kernel 1: per-row inverse norms ----------------
// One wave per feature row (10*2048 rows of 128 floats).
__global__ __launch_bounds__(256) void prep_norms(const float* __restrict__ preds,
                                                  float* __restrict__ inv_norm) {
  int wave = threadIdx.x >> 5;
  int lane = threadIdx.x & 31;
  int row  = blockIdx.x * 8 + wave;             // 0 .. 20479
  const float4* p = (const float4*)(preds + (size_t)row * DIM);
  float4 v = p[lane];
  float s = v.x * v.x + v.y * v.y + v.z * v.z + v.w * v.w;
#pragma unroll
  for (int m = 16; m >= 1; m >>= 1) s += __shfl_xor(s, m, 32);
  if (lane == 0) inv_norm[row] = rsqrtf(s);
}

// ---------------- Kernel 2: per-class positive counts ----------------
__global__ void count_pos(const int* __restrict__ target, int* __restrict__ pcnt) {
  __shared__ int cnt[NUM_CLASSES];
  if (threadIdx.x < NUM_CLASSES) cnt[threadIdx.x] = 0;
  __syncthreads();
  for (int i = threadIdx.x; i < BATCH; i += 256) {
    int t = target[i];
    if (t >= 0 && t < NUM_CLASSES) atomicAdd(&cnt[t], 1);
  }
  __syncthreads();
  if (threadIdx.x < NUM_CLASSES) pcnt[threadIdx.x] = cnt[threadIdx.x];
}

// ---------------- Kernel 3: fused cosine-GEMM + masked softmax stats ----------------
// Grid: 10 classes x 64 row-blocks (32 rows). 256 threads = 8 waves, each wave a
// 16x16 WMMA tile; wave grid is 2 (rows) x 4 (cols) -> 32x64 macro tile per N-step.
__global__ __launch_bounds__(256) void supcon_main(const float* __restrict__ preds,
                                                   const int* __restrict__ target,
                                                   const float* __restrict__ log_vars,
                                                   const float* __restrict__ inv_norm,
                                                   const int* __restrict__ pcnt,
                                                   float* __restrict__ partial) {
  __shared__ float Alds[MTILE * LDS_STRIDE];     // 16.9 KB, rows pre-scaled by 1/||f||
  __shared__ float Blds[NTILE * LDS_STRIDE];     // 33.8 KB
  __shared__ unsigned char posf[BATCH];          // 2 KB: target==class flags
  __shared__ float wsZ[4][MTILE];                // per-(col-quarter) row partials
  __shared__ float wsS[4][MTILE];

  const int cls = blockIdx.x / (BATCH / MTILE);
  const int rb  = blockIdx.x % (BATCH / MTILE);
  const int a0  = rb * MTILE;
  const int tid  = threadIdx.x;
  const int lane = tid & 31, wave = tid >> 5;
  const int wr = wave & 1, wc = wave >> 1;       // wave's row-half / col-quarter
  const int r = lane & 15, h = lane >> 4;        // fragment lane decomposition

  const float* __restrict__ Fc  = preds + (size_t)cls * BATCH * DIM;
  const float* __restrict__ inv = inv_norm + cls * BATCH;

  // ---- stage positive flags + A tile (scaled) ----
  for (int i = tid; i < BATCH; i += 256) posf[i] = (target[i] == cls) ? 1 : 0;
#pragma unroll
  for (int i = 0; i < 4; i++) {                  // 32 rows * 32 float4 = 1024
    int idx = tid + i * 256;
    int row = idx >> 5;
    int k4  = idx & 31;
    int g = a0 + row;
    float sc = inv[g];
    float4 v = *(const float4*)(Fc + (size_t)g * DIM + k4 * 4);
    float* dst = &Alds[row * LDS_STRIDE + k4 * 4];
    dst[0] = v.x * sc; dst[1] = v.y * sc; dst[2] = v.z * sc; dst[3] = v.w * sc;
  }
  __syncthreads();

  // ---- preload A fragments: A[m][4*kk + 2h + {0,1}], m = wr*16 + r ----
  v2f afrag[32];
  {
    const float* Arow = &Alds[(wr * 16 + r) * LDS_STRIDE];
#pragma unroll
    for (int kk = 0; kk < 32; kk++)
      afrag[kk] = *(const v2f*)&Arow[kk * 4 + 2 * h];
  }
  bool posa[8];
#pragma unroll
  for (int j = 0; j < 8; j++) posa[j] = posf[a0 + wr * 16 + 8 * h + j] != 0;

  // ---- software-pipelined B staging regs ----
  float4 bv[8]; float bsc[8];
  {
#pragma unroll
    for (int i = 0; i < 8; i++) {                // 64 rows * 32 float4 = 2048
      int idx = tid + i * 256;
      int row = idx >> 5;
      int k4  = idx & 31;
      int g = 0 * NTILE + row;
      bv[i]  = *(const float4*)(Fc + (size_t)g * DIM + k4 * 4);
      bsc[i] = inv[g];
    }
  }

  float zpart[8], spart[8];
#pragma unroll
  for (int j = 0; j < 8; j++) { zpart[j] = 0.f; spart[j] = 0.f; }

  const float invT = 1.0f / TEMP;

  for (int s = 0; s < NSTEPS; s++) {
    __syncthreads();                             // previous tile's LDS reads done
#pragma unroll
    for (int i = 0; i < 8; i++) {                // commit staged B (scaled)
      int idx = tid + i * 256;
      int row = idx >> 5;
      int k4  = idx & 31;
      float* dst = &Blds[row * LDS_STRIDE + k4 * 4];
      float sc = bsc[i];
      dst[0] = bv[i].x * sc; dst[1] = bv[i].y * sc;
      dst[2] = bv[i].z * sc; dst[3] = bv[i].w * sc;
    }
    __syncthreads();                             // B tile visible
    if (s + 1 < NSTEPS) {                        // prefetch next tile during compute
      int nb = (s + 1) * NTILE;
#pragma unroll
      for (int i = 0; i < 8; i++) {
        int idx = tid + i * 256;
        int row = idx >> 5;
        int k4  = idx & 31;
        int g = nb + row;
        bv[i]  = *(const float4*)(Fc + (size_t)g * DIM + k4 * 4);
        bsc[i] = inv[g];
      }
    }

    // ---- 16x16 cosine tile via 32 x V_WMMA_F32_16X16X4_F32 ----
    v8f acc = {0.f, 0.f, 0.f, 0.f, 0.f, 0.f, 0.f, 0.f};
    const float* Brow = &Blds[(wc * 16 + r) * LDS_STRIDE];
#pragma unroll
    for (int kk = 0; kk < 32; kk++) {
      v2f b = *(const v2f*)&Brow[kk * 4 + 2 * h];
      acc = __builtin_amdgcn_wmma_f32_16x16x4_f32(false, afrag[kk], false, b,
                                                  (short)0, acc, false, false);
    }

    // ---- fold tile into running softmax stats ----
    int gcol = s * NTILE + wc * 16 + r;          // lane's column
    bool posb = posf[gcol] != 0;
#pragma unroll
    for (int j = 0; j < 8; j++) {
      int grow = a0 + wr * 16 + 8 * h + j;       // C layout: VGPR j, half h
      float logit = acc[j] * invT;
      bool diag = (gcol == grow);
      zpart[j] += diag ? 0.f : __expf(logit);
      spart[j] += ((posa[j] == posb) && !diag) ? logit : 0.f;
    }
  }

  // ---- reduce columns: 16 lanes per half share the same rows ----
#pragma unroll
  for (int j = 0; j < 8; j++) {
    float z = zpart[j], sm = spart[j];
#pragma unroll
    for (int m = 8; m >= 1; m >>= 1) {           // masks<16: halves stay separate
      z  += __shfl_xor(z,  m, 32);
      sm += __shfl_xor(sm, m, 32);
    }
    if (r == 0) {                                // lane 0 / lane 16 of each wave
      wsZ[wc][wr * 16 + 8 * h + j] = z;          // fixed slots -> deterministic
      wsS[wc][wr * 16 + 8 * h + j] = sm;
    }
  }
  __syncthreads();

  // ---- per-row loss, block partial ----
  if (tid < MTILE) {
    int ga = a0 + tid;
    float Z = wsZ[0][tid] + wsZ[1][tid] + wsZ[2][tid] + wsZ[3][tid];
    float S = wsS[0][tid] + wsS[1][tid] + wsS[2][tid] + wsS[3][tid];
    float P = (float)pcnt[cls];
    float cnt = (posf[ga] ? P : ((float)BATCH - P)) - 1.0f;
    float val = S / cnt - __logf(Z);
    float contrib = -(TEMP / BASE_TEMP) * val * (1.0f / (float)BATCH);
#pragma unroll
    for (int m = 16; m >= 1; m >>= 1) contrib += __shfl_xor(contrib, m, 32);
    if (tid == 0) partial[blockIdx.x] = __expf(-log_vars[cls]) * contrib;
  }
}

// ---------------- Kernel 4: deterministic final reduction ----------------
__global__ void final_reduce(const float* __restrict__ partial,
                             const float* __restrict__ log_vars,
                             float* __restrict__ out, int n) {
  __shared__ float sh[256];
  float s = 0.f;
  for (int i = threadIdx.x; i < n; i += 256) s += partial[i];
  sh[threadIdx.x] = s;
  __syncthreads();
  for (int m = 128; m >= 1; m >>= 1) {
    if (threadIdx.x < m) sh[threadIdx.x] += sh[threadIdx.x + m];
    __syncthreads();
  }
  if (threadIdx.x == 0) {
    float lv = 0.f;
    for (int c = 0; c < NUM_CLASSES; c++) lv += log_vars[c];
    out[0] = sh[0] + lv;
  }
}

extern "C" void kernel_launch(void* const* d_in, const int* in_sizes, int n_in,
                              void* d_out, int out_size, void* d_ws, size_t ws_size,
                              hipStream_t stream) {
  const float* preds    = (const float*)d_in[0];   // [10,2048,128] f32
  const int*   target   = (const int*)d_in[1];     // [2048] int
  const float* log_vars = (const float*)d_in[2];   // [10] f32
  float* out = (float*)d_out;

  float* inv_norm = (float*)d_ws;                          // 20480 f32
  float* partial  = inv_norm + NUM_CLASSES * BATCH;        // 640 f32
  int*   pcnt     = (int*)(partial + NBLOCKS);             // 10 i32

  prep_norms<<<NUM_CLASSES * BATCH / 8, 256, 0, stream>>>(preds, inv_norm);
  count_pos<<<1, 256, 0, stream>>>(target, pcnt);
  supcon_main<<<NBLOCKS, 256, 0, stream>>>(preds, target, log_vars,
                                           inv_norm, pcnt, partial);
  final_reduce<<<1, 256, 0, stream>>>(partial, log_vars, out, NBLOCKS);
}